// OlmoSampler_67800353734695
// MI455X (gfx1250) — compile-verified
//
#include <hip/hip_runtime.h>
#include <stdint.h>

#define TPB 1024
#define NW (TPB / 32)
#define MAXL 512

typedef unsigned int v4u __attribute__((ext_vector_type(4)));
typedef int          v8i __attribute__((ext_vector_type(8)));
typedef int          v4i __attribute__((ext_vector_type(4)));

__device__ __forceinline__ float blk_max(float v, float* red) {
    int tid = threadIdx.x;
    red[tid] = v; __syncthreads();
    for (int s = TPB / 2; s > 0; s >>= 1) {
        if (tid < s) red[tid] = fmaxf(red[tid], red[tid + s]);
        __syncthreads();
    }
    float r = red[0]; __syncthreads();
    return r;
}

__device__ __forceinline__ float blk_sum(float v, float* red) {
    int tid = threadIdx.x;
    red[tid] = v; __syncthreads();
    for (int s = TPB / 2; s > 0; s >>= 1) {
        if (tid < s) red[tid] += red[tid + s];
        __syncthreads();
    }
    float r = red[0]; __syncthreads();
    return r;
}

__global__ __launch_bounds__(TPB) void OlmoSampler_67800353734695_kernel(
    const float* __restrict__ logits,
    const long long* __restrict__ tokens,
    const float* __restrict__ presence,
    const float* __restrict__ freq,
    const float* __restrict__ temps,
    const float* __restrict__ toppv,
    const long long* __restrict__ topkv,
    float* __restrict__ out,
    int V, int L)
{
    __shared__ long long s_tok[MAXL];
    __shared__ int       s_ftok[MAXL];
    __shared__ float     s_fdel[MAXL];
    __shared__ unsigned  s_cnt[NW][256];   // per-wave privatized histograms
    __shared__ float     s_sum[NW][256];
    __shared__ float     s_red[TPB];
    __shared__ unsigned  s_wsum[NW];
    __shared__ unsigned  s_prefix, s_C, s_cnteq, s_base;
    __shared__ float     s_Sc;

    const int row = blockIdx.x;
    const int tid = threadIdx.x;
    const int wid = tid >> 5;
    const float* __restrict__ xrow = logits + (size_t)row * V;
    float* __restrict__ orow = out + (size_t)row * V;
    const float4* __restrict__ xrow4 = reinterpret_cast<const float4*>(xrow);
    float4* __restrict__ orow4 = reinterpret_cast<float4*>(orow);
    const int V4 = V >> 2;
    const int Vt = V4 << 2;   // start of scalar tail

    const float invT = 1.0f / temps[row];
    const float fp   = freq[row];
    const float pp   = presence[row];
    const float topp = toppv[row];
    const int   topk = (int)topkv[row];

    // ---- TDM: DMA this row's output tokens (L x int64) into LDS (wave 0 only) ----
    if (tid < 32) {
        uint64_t ga  = (uint64_t)(uintptr_t)(tokens + (size_t)row * L);
        uint32_t lda = (uint32_t)(uintptr_t)(void*)s_tok;   // low 32 bits = LDS offset
        uint32_t d0  = (uint32_t)L;
        v4u g0;
        g0[0] = 1u;                                          // count=1, is_restore=0
        g0[1] = lda;                                         // lds_addr
        g0[2] = (uint32_t)(ga & 0xffffffffu);                // global_addr[31:0]
        g0[3] = (uint32_t)((ga >> 32) & 0x01ffffffu) | (2u << 30); // addr[56:32] | type=2
        v8i g1;
        g1[0] = (int)(3u << 16);                             // data_size=3 (8B), mask=0
        g1[1] = (int)((d0 & 0xffffu) << 16);                 // tensor_dim0 lo16
        g1[2] = (int)(((d0 >> 16) & 0xffffu) | (1u << 16));  // dim0 hi | tensor_dim1=1 lo
        g1[3] = (int)((d0 & 0xffffu) << 16);                 // dim1 hi=0 | tile_dim0=L
        g1[4] = (int)1u;                                     // tile_dim1=1, tile_dim2=0
        g1[5] = (int)d0;                                     // tensor_dim0_stride[31:0]
        g1[6] = 0;
        g1[7] = 0;
        v4i gz4 = {0, 0, 0, 0};
        v8i gz8 = {0, 0, 0, 0, 0, 0, 0, 0};
        __builtin_amdgcn_tensor_load_to_lds(g0, g1, gz4, gz4, gz8, 0);
        __builtin_amdgcn_s_wait_tensorcnt(0);
    }
    __syncthreads();

    // ---- per-token counts + first occurrence (stable dedupe) ----
    for (int j = tid; j < L; j += TPB) {
        int t = (int)s_tok[j];
        int cnt = 0; bool first = true;
        for (int i = 0; i < L; ++i) {
            int ti = (int)s_tok[i];
            cnt += (ti == t);
            if (ti == t && i < j) first = false;
        }
        s_ftok[j] = first ? t : -1;
        s_fdel[j] = fp * (float)cnt + pp;
    }
    __syncthreads();

    // ---- copy logits to out (b128), then scatter sparse penalties ----
    for (int i = tid; i < V4; i += TPB) orow4[i] = xrow4[i];
    for (int v = Vt + tid; v < V; v += TPB) orow[v] = xrow[v];
    __threadfence_block();
    __syncthreads();
    for (int j = tid; j < L; j += TPB) {
        int t = s_ftok[j];
        if (t >= 0) orow[t] -= s_fdel[j];
    }
    __threadfence_block();
    __syncthreads();

    // ---- pass 1: row max of y = x/T ----
    float lmax = -3.402823466e38f;
    for (int i = tid; i < V4; i += TPB) {
        float4 a = orow4[i];
        lmax = fmaxf(lmax, fmaxf(fmaxf(a.x, a.y), fmaxf(a.z, a.w)) * invT);
    }
    for (int v = Vt + tid; v < V; v += TPB) lmax = fmaxf(lmax, orow[v] * invT);
    float m = blk_max(lmax, s_red);

    // ---- pass 2: softmax denom, then write probs p into orow ----
    float lsum = 0.f;
    for (int i = tid; i < V4; i += TPB) {
        float4 a = orow4[i];
        lsum += expf(a.x * invT - m) + expf(a.y * invT - m)
              + expf(a.z * invT - m) + expf(a.w * invT - m);
    }
    for (int v = Vt + tid; v < V; v += TPB) lsum += expf(orow[v] * invT - m);
    float S = blk_sum(lsum, s_red);
    float invS = 1.0f / S;
    for (int i = tid; i < V4; i += TPB) {
        float4 a = orow4[i];
        a.x = expf(a.x * invT - m) * invS;
        a.y = expf(a.y * invT - m) * invS;
        a.z = expf(a.z * invT - m) * invS;
        a.w = expf(a.w * invT - m) * invS;
        orow4[i] = a;
    }
    for (int v = Vt + tid; v < V; v += TPB) orow[v] = expf(orow[v] * invT - m) * invS;
    __threadfence_block();
    __syncthreads();

    // ---- 4-pass radix select on prob bit pattern (nonneg => uint order) ----
    if (tid == 0) { s_prefix = 0u; s_C = 0u; s_Sc = 0.f; s_cnteq = 0u; }
    __syncthreads();
    for (int pass = 0; pass < 4; ++pass) {
        const int shift = 24 - 8 * pass;
        for (int b = tid; b < NW * 256; b += TPB) {
            (&s_cnt[0][0])[b] = 0u;
            (&s_sum[0][0])[b] = 0.f;
        }
        __syncthreads();
        const unsigned pref = s_prefix;
        for (int i = tid; i < V4; i += TPB) {
            float4 a = orow4[i];
            float pc[4] = {a.x, a.y, a.z, a.w};
            #pragma unroll
            for (int c = 0; c < 4; ++c) {
                unsigned u = __float_as_uint(pc[c]);
                bool match = (pass == 0) || ((u >> (shift + 8)) == pref);
                if (match) {
                    unsigned b = (u >> shift) & 255u;
                    atomicAdd(&s_cnt[wid][b], 1u);
                    atomicAdd(&s_sum[wid][b], pc[c]);
                }
            }
        }
        for (int v = Vt + tid; v < V; v += TPB) {
            float p = orow[v];
            unsigned u = __float_as_uint(p);
            bool match = (pass == 0) || ((u >> (shift + 8)) == pref);
            if (match) {
                unsigned b = (u >> shift) & 255u;
                atomicAdd(&s_cnt[wid][b], 1u);
                atomicAdd(&s_sum[wid][b], p);
            }
        }
        __syncthreads();
        // deterministic cross-copy reduction into copy 0
        for (int b = tid; b < 256; b += TPB) {
            unsigned ct = 0; float sm = 0.f;
            for (int w = 0; w < NW; ++w) { ct += s_cnt[w][b]; sm += s_sum[w][b]; }
            s_cnt[0][b] = ct; s_sum[0][b] = sm;
        }
        __syncthreads();
        if (tid == 0) {
            unsigned C = s_C; float Sc = s_Sc;
            int choose = -1, lastb = -1;
            unsigned lastC = C; float lastSc = Sc;
            for (int b = 255; b >= 0; --b) {
                unsigned c = s_cnt[0][b];
                if (!c) continue;
                float sm = s_sum[0][b];
                if ((C + c >= (unsigned)topk) || (Sc + sm > topp)) { choose = b; break; }
                lastb = b; lastC = C; lastSc = Sc;
                C += c; Sc += sm;
            }
            if (choose < 0) { choose = lastb; C = lastC; Sc = lastSc; }
            s_C = C; s_Sc = Sc;
            s_prefix = (pref << 8) | (unsigned)choose;
            if (pass == 3) s_cnteq = s_cnt[0][choose];
        }
        __syncthreads();
    }

    const unsigned t_bits = s_prefix;
    const float    p_t    = __uint_as_float(t_bits);
    const float    s_gt   = s_Sc;
    const int      cnt_eq = (int)s_cnteq;

    // tie budgets: by top-k and by top-p (exclusive cumsum s_gt + j*p_t <= topp)
    int r_k = topk - (int)s_C;
    int r_p;
    if (p_t > 0.f) {
        float budget = topp - s_gt;
        if (budget < 0.f) r_p = 0;
        else {
            float q = budget / p_t;
            r_p = (q >= (float)cnt_eq) ? cnt_eq : ((int)q + 1);
        }
    } else {
        r_p = cnt_eq;
    }
    int r_tie = min(cnt_eq, min(r_k, r_p));
    if (r_tie < 0) r_tie = 0;

    // ---- exact kept-mass (tree reduction, deterministic) ----
    float ls = 0.f;
    for (int i = tid; i < V4; i += TPB) {
        float4 a = orow4[i];
        if (__float_as_uint(a.x) > t_bits) ls += a.x;
        if (__float_as_uint(a.y) > t_bits) ls += a.y;
        if (__float_as_uint(a.z) > t_bits) ls += a.z;
        if (__float_as_uint(a.w) > t_bits) ls += a.w;
    }
    for (int v = Vt + tid; v < V; v += TPB) {
        float p = orow[v];
        if (__float_as_uint(p) > t_bits) ls += p;
    }
    float S_keep = blk_sum(ls, s_red) + (float)r_tie * p_t;
    float invK = 1.0f / S_keep;

    // ---- final write ----
    if (cnt_eq <= r_tie) {
        // all ties kept: simple threshold pass
        for (int i = tid; i < V4; i += TPB) {
            float4 a = orow4[i];
            a.x = (__float_as_uint(a.x) >= t_bits) ? a.x * invK : 0.f;
            a.y = (__float_as_uint(a.y) >= t_bits) ? a.y * invK : 0.f;
            a.z = (__float_as_uint(a.z) >= t_bits) ? a.z * invK : 0.f;
            a.w = (__float_as_uint(a.w) >= t_bits) ? a.w * invK : 0.f;
            orow4[i] = a;
        }
        for (int v = Vt + tid; v < V; v += TPB) {
            float p = orow[v];
            orow[v] = (__float_as_uint(p) >= t_bits) ? p * invK : 0.f;
        }
    } else {
        // rank ties by ascending index (matches stable argsort) via block scan
        if (tid == 0) s_base = 0u;
        __syncthreads();
        const int lane = tid & 31;
        for (int c0 = 0; c0 < V; c0 += TPB) {
            int v = c0 + tid;
            float p = (v < V) ? orow[v] : -1.f;
            unsigned u = __float_as_uint(p);
            bool q = (v < V) && (u == t_bits);
            unsigned long long bal = __ballot(q);
            unsigned wcount  = (unsigned)__popcll(bal);
            unsigned wprefix = (unsigned)__popcll(bal & ((1ull << lane) - 1ull));
            if (lane == 0) s_wsum[wid] = wcount;
            __syncthreads();
            unsigned wbase = 0, total = 0;
            for (int w2 = 0; w2 < NW; ++w2) {
                unsigned c = s_wsum[w2];
                if (w2 < wid) wbase += c;
                total += c;
            }
            unsigned base = s_base;
            __syncthreads();
            if (tid == 0) s_base = base + total;
            if (v < V) {
                bool keep;
                if (u > t_bits)      keep = true;
                else if (q)          keep = ((int)(base + wbase + wprefix) < r_tie);
                else                 keep = false;
                orow[v] = keep ? p * invK : 0.f;
            }
            __syncthreads();
        }
    }
}

extern "C" void kernel_launch(void* const* d_in, const int* in_sizes, int n_in,
                              void* d_out, int out_size, void* d_ws, size_t ws_size,
                              hipStream_t stream) {
    (void)n_in; (void)d_ws; (void)ws_size; (void)out_size;
    const float*     logits   = (const float*)d_in[0];
    const long long* tokens   = (const long long*)d_in[1];
    const float*     presence = (const float*)d_in[2];
    const float*     freq     = (const float*)d_in[3];
    const float*     temps    = (const float*)d_in[4];
    const float*     topps    = (const float*)d_in[5];
    const long long* topks    = (const long long*)d_in[6];

    const int B = in_sizes[2];
    const int V = in_sizes[0] / B;
    int L = in_sizes[1] / B;
    if (L > MAXL) L = MAXL;

    OlmoSampler_67800353734695_kernel<<<B, TPB, 0, stream>>>(
        logits, tokens, presence, freq, temps, topps, topks, (float*)d_out, V, L);
}